// GCN_82944408420994
// MI455X (gfx1250) — compile-verified
//
#include <hip/hip_runtime.h>
#include <math.h>

typedef __attribute__((ext_vector_type(2))) float v2f;
typedef __attribute__((ext_vector_type(8))) float v8f;

#define NNODES 50000
#define NEDGES 800000
#define NGRAPH 256
#define ADDF   7
#define BNEPS  1e-5f

__device__ __forceinline__ float gelu_exact(float x) {
    return 0.5f * x * (1.0f + erff(x * 0.7071067811865475f));
}

// ---------------- utility ----------------
__global__ void gcn_fill(float* __restrict__ p, float v, int n) {
    int i = blockIdx.x * blockDim.x + threadIdx.x;
    if (i < n) p[i] = v;
}

// deg[dst] += 1 for every edge (deg pre-filled with 1.0 for self loops)
__global__ void gcn_deg(const int* __restrict__ ei, float* __restrict__ deg) {
    int e = blockIdx.x * blockDim.x + threadIdx.x;
    if (e < NEDGES) atomicAdd(&deg[ei[NEDGES + e]], 1.0f);
}

// in place: dis = rsqrt(deg)   (deg >= 1 always, self loops included)
__global__ void gcn_dis(float* __restrict__ d) {
    int i = blockIdx.x * blockDim.x + threadIdx.x;
    if (i < NNODES) d[i] = rsqrtf(d[i]);
}

// ---------------- layer-1 linear (Cin = 1) ----------------
__global__ void gcn_lin1(const float* __restrict__ x, const float* __restrict__ W,
                         float* __restrict__ h) {
    int i = blockIdx.x * blockDim.x + threadIdx.x;   // N*16
    if (i < NNODES * 16) {
        int n = i >> 4, c = i & 15;
        h[i] = x[n] * W[c];
    }
}

// ---------------- WMMA fp32 GEMM: Y[N,Cout] = X[N,K] @ W[K,Cout] ----------------
// One wave (32 threads) per 16x16 output tile; K stepped by 4 with
// V_WMMA_F32_16X16X4_F32. N = 50000 = 3125*16 (exact), K,Cout multiples of 4/16.
__global__ void gcn_wmma_gemm(const float* __restrict__ X, const float* __restrict__ W,
                              float* __restrict__ Y, int K, int Cout) {
    const int lane = threadIdx.x;          // 0..31, EXEC all ones
    const int mt   = blockIdx.x << 4;      // row tile base
    const int nt   = blockIdx.y << 4;      // col tile base
    const int ml   = lane & 15;            // row (A) / col (B) within tile
    const int ksel = (lane >> 4) << 1;     // 0 for lanes 0-15, 2 for 16-31

    v8f acc = {};
    for (int k0 = 0; k0 < K; k0 += 4) {
        v2f a, b;
        // A fragment 16x4: lane ml holds row mt+ml, K = k0+ksel, k0+ksel+1
        a.x = X[(mt + ml) * K + k0 + ksel];
        a.y = X[(mt + ml) * K + k0 + ksel + 1];
        // B fragment 4x16: lane ml holds col nt+ml, K = k0+ksel, k0+ksel+1
        b.x = W[(k0 + ksel) * Cout + nt + ml];
        b.y = W[(k0 + ksel + 1) * Cout + nt + ml];
        acc = __builtin_amdgcn_wmma_f32_16x16x4_f32(
            /*neg_a=*/false, a, /*neg_b=*/false, b,
            /*c_mod=*/(short)0, acc, /*reuse_a=*/false, /*reuse_b=*/false);
    }
    // C/D layout: VGPR v -> M = v (lanes 0-15) or v+8 (lanes 16-31), N = lane&15
    const int col = nt + ml;
    const int row = mt + ((lane >> 4) << 3);
#pragma unroll
    for (int v = 0; v < 8; ++v)
        Y[(row + v) * Cout + col] = acc[v];
}

// ---------------- edge scatter: agg[dst] += h[src] * dis[src]*dis[dst] ----------------
__global__ void gcn_scatter(const int* __restrict__ ei, const float* __restrict__ dis,
                            const float* __restrict__ h, float* __restrict__ agg, int C) {
    int i = blockIdx.x * blockDim.x + threadIdx.x;   // E * C
    if (i < NEDGES * C) {
        int e = i / C, c = i - e * C;
        int s = ei[e], d = ei[NEDGES + e];
        float nrm = dis[s] * dis[d];
        atomicAdd(&agg[d * C + c], h[s * C + c] * nrm);
    }
}

// self loops: agg[n] += h[n] * dis[n]^2   (separate kernel, no atomics needed)
__global__ void gcn_selfloop(const float* __restrict__ dis, const float* __restrict__ h,
                             float* __restrict__ agg, int C) {
    int i = blockIdx.x * blockDim.x + threadIdx.x;   // N * C
    if (i < NNODES * C) {
        int n = i / C;
        float w = dis[n];
        agg[i] += h[i] * w * w;
    }
}

// out = gelu(agg + bias[c])
__global__ void gcn_bias_gelu(const float* __restrict__ agg, const float* __restrict__ b,
                              float* __restrict__ out, int C) {
    int i = blockIdx.x * blockDim.x + threadIdx.x;
    if (i < NNODES * C) {
        int c = i % C;
        out[i] = gelu_exact(agg[i] + b[c]);
    }
}

// ---------------- BatchNorm (training mode, biased variance) ----------------
__global__ void gcn_bn_stats(const float* __restrict__ x, float* __restrict__ stats, int C) {
    int c = blockIdx.x;
    float s = 0.f, s2 = 0.f;
    for (int n = threadIdx.x; n < NNODES; n += blockDim.x) {
        float v = x[n * C + c];
        s += v; s2 += v * v;
    }
    __shared__ float sh[256], sh2[256];
    sh[threadIdx.x] = s; sh2[threadIdx.x] = s2;
    __syncthreads();
    for (int o = blockDim.x >> 1; o > 0; o >>= 1) {
        if (threadIdx.x < (unsigned)o) {
            sh[threadIdx.x]  += sh[threadIdx.x + o];
            sh2[threadIdx.x] += sh2[threadIdx.x + o];
        }
        __syncthreads();
    }
    if (threadIdx.x == 0) {
        float mu  = sh[0] * (1.0f / NNODES);
        float var = sh2[0] * (1.0f / NNODES) - mu * mu;
        stats[c] = mu;
        stats[C + c] = var;
    }
}

__global__ void gcn_bn_apply(float* __restrict__ x, const float* __restrict__ stats,
                             const float* __restrict__ g, const float* __restrict__ be, int C) {
    int i = blockIdx.x * blockDim.x + threadIdx.x;
    if (i < NNODES * C) {
        int c = i % C;
        float mu = stats[c], var = stats[C + c];
        x[i] = g[c] * (x[i] - mu) * rsqrtf(var + BNEPS) + be[c];
    }
}

// ---------------- global mean pool ----------------
__global__ void gcn_pool_sum(const float* __restrict__ h, const int* __restrict__ batch,
                             float* __restrict__ pooled) {
    int i = blockIdx.x * blockDim.x + threadIdx.x;   // N * 128
    if (i < NNODES * 128) {
        int n = i >> 7, c = i & 127;
        atomicAdd(&pooled[batch[n] * 128 + c], h[i]);
    }
}

__global__ void gcn_pool_cnt(const int* __restrict__ batch, float* __restrict__ cnt) {
    int n = blockIdx.x * blockDim.x + threadIdx.x;
    if (n < NNODES) atomicAdd(&cnt[batch[n]], 1.0f);
}

// ---------------- fused MLP head: 135 -> 128 -> 64 -> 32 -> 2 (sigmoid) ----------------
__global__ void gcn_mlp(const float* __restrict__ pooled, const float* __restrict__ cnt,
                        const float* __restrict__ yex,
                        const float* __restrict__ lw1, const float* __restrict__ lb1,
                        const float* __restrict__ lw2, const float* __restrict__ lb2,
                        const float* __restrict__ lw3, const float* __restrict__ lb3,
                        const float* __restrict__ lw4, const float* __restrict__ lb4,
                        float* __restrict__ out) {
    __shared__ float z0[135], z1[128], z2[64], z3[32];
    const int g = blockIdx.x, t = threadIdx.x;        // 128 threads
    float invc = 1.0f / fmaxf(cnt[g], 1.0f);
    z0[t] = pooled[g * 128 + t] * invc;
    if (t < ADDF) z0[128 + t] = yex[g * ADDF + t];
    __syncthreads();
    {   // 135 -> 128
        float a = lb1[t];
        for (int k = 0; k < 135; ++k) a += z0[k] * lw1[k * 128 + t];
        z1[t] = gelu_exact(a);
    }
    __syncthreads();
    if (t < 64) {   // 128 -> 64
        float a = lb2[t];
        for (int k = 0; k < 128; ++k) a += z1[k] * lw2[k * 64 + t];
        z2[t] = gelu_exact(a);
    }
    __syncthreads();
    if (t < 32) {   // 64 -> 32
        float a = lb3[t];
        for (int k = 0; k < 64; ++k) a += z2[k] * lw3[k * 32 + t];
        z3[t] = gelu_exact(a);
    }
    __syncthreads();
    if (t < 2) {    // 32 -> 2, sigmoid
        float a = lb4[t];
        for (int k = 0; k < 32; ++k) a += z3[k] * lw4[k * 2 + t];
        out[g * 2 + t] = 1.0f / (1.0f + expf(-a));
    }
}

// ---------------- host side ----------------
static inline int cdiv(long a, int b) { return (int)((a + b - 1) / b); }

extern "C" void kernel_launch(void* const* d_in, const int* in_sizes, int n_in,
                              void* d_out, int out_size, void* d_ws, size_t ws_size,
                              hipStream_t stream) {
    const float* x    = (const float*)d_in[0];
    const int*   ei   = (const int*)d_in[1];
    const int*   batch= (const int*)d_in[2];
    const float* yex  = (const float*)d_in[3];
    const float* W1 = (const float*)d_in[4],  *b1 = (const float*)d_in[5];
    const float* W2 = (const float*)d_in[6],  *b2 = (const float*)d_in[7];
    const float* W3 = (const float*)d_in[8],  *b3 = (const float*)d_in[9];
    const float* W4 = (const float*)d_in[10], *b4 = (const float*)d_in[11];
    const float* g1 = (const float*)d_in[12], *be1 = (const float*)d_in[13];
    const float* g2 = (const float*)d_in[14], *be2 = (const float*)d_in[15];
    const float* g3 = (const float*)d_in[16], *be3 = (const float*)d_in[17];
    const float* lw1 = (const float*)d_in[18], *lb1 = (const float*)d_in[19];
    const float* lw2 = (const float*)d_in[20], *lb2 = (const float*)d_in[21];
    const float* lw3 = (const float*)d_in[22], *lb3 = (const float*)d_in[23];
    const float* lw4 = (const float*)d_in[24], *lb4 = (const float*)d_in[25];
    float* out = (float*)d_out;

    const long N128 = (long)NNODES * 128;
    float* ws     = (float*)d_ws;
    float* bufH   = ws;                 // GEMM output      [N,128]
    float* bufAgg = ws + N128;          // scatter target   [N,128]
    float* bufX   = ws + 2 * N128;      // activations      [N,128]
    float* dis    = ws + 3 * N128;      // deg -> rsqrt(deg) [N]
    float* stats  = dis + NNODES;       // BN mean/var      [256]
    float* pooled = stats + 256;        // pool sums        [G,128]
    float* cnt    = pooled + (long)NGRAPH * 128;  // node counts [G]

    const int T = 256;

    // degree / normalization (self loops via deg init = 1)
    gcn_fill<<<cdiv(NNODES, T), T, 0, stream>>>(dis, 1.0f, NNODES);
    gcn_deg<<<cdiv(NEDGES, T), T, 0, stream>>>(ei, dis);
    gcn_dis<<<cdiv(NNODES, T), T, 0, stream>>>(dis);

    struct { const float *W, *b, *g, *be; int Cin, Cout; } L[4] = {
        { W1, b1, g1, be1,  1,  16 },
        { W2, b2, g2, be2, 16,  32 },
        { W3, b3, g3, be3, 32,  64 },
        { W4, b4, g3, be3, 64, 128 },   // layer 4: no BN (g/be unused)
    };

    for (int l = 0; l < 4; ++l) {
        const int C = L[l].Cout;
        const long NC = (long)NNODES * C;
        // 1) dense GEMM  h = x_in @ W
        if (l == 0) {
            gcn_lin1<<<cdiv(NC, T), T, 0, stream>>>(x, L[l].W, bufH);
        } else {
            dim3 grid(NNODES / 16, C / 16);
            gcn_wmma_gemm<<<grid, 32, 0, stream>>>(bufX, L[l].W, bufH, L[l].Cin, C);
        }
        // 2) normalized scatter-add + self loops
        gcn_fill<<<cdiv(NC, T), T, 0, stream>>>(bufAgg, 0.0f, (int)NC);
        gcn_scatter<<<cdiv((long)NEDGES * C, T), T, 0, stream>>>(ei, dis, bufH, bufAgg, C);
        gcn_selfloop<<<cdiv(NC, T), T, 0, stream>>>(dis, bufH, bufAgg, C);
        // 3) bias + GELU
        gcn_bias_gelu<<<cdiv(NC, T), T, 0, stream>>>(bufAgg, L[l].b, bufX, C);
        // 4) BatchNorm (layers 1-3 only)
        if (l < 3) {
            gcn_bn_stats<<<C, 256, 0, stream>>>(bufX, stats, C);
            gcn_bn_apply<<<cdiv(NC, T), T, 0, stream>>>(bufX, stats, L[l].g, L[l].be, C);
        }
    }

    // global mean pool + MLP head
    gcn_fill<<<cdiv((long)NGRAPH * 128, T), T, 0, stream>>>(pooled, 0.0f, NGRAPH * 128);
    gcn_fill<<<cdiv(NGRAPH, T), T, 0, stream>>>(cnt, 0.0f, NGRAPH);
    gcn_pool_sum<<<cdiv(N128, T), T, 0, stream>>>(bufX, batch, pooled);
    gcn_pool_cnt<<<cdiv(NNODES, T), T, 0, stream>>>(batch, cnt);
    gcn_mlp<<<NGRAPH, 128, 0, stream>>>(pooled, cnt, yex,
                                        lw1, lb1, lw2, lb2, lw3, lb3, lw4, lb4, out);
}